// FairGMM_14508399526602
// MI455X (gfx1250) — compile-verified
//
#include <hip/hip_runtime.h>
#include <hip/hip_bf16.h>
#include <math.h>

// ---------------------------------------------------------------------------
// Slot-attention GMM forward for MI455X (gfx1250, wave32).
// All dense matmuls run on V_WMMA_F32_16X16X4_F32 (exact f32, 16x16 tiles).
// Problem is tiny (~2.4 GFLOP, 17MB traffic): latency bound, so FP32 WMMA is
// the right precision; k (16.8MB) stays resident in the 192MB L2 across the
// 4 iterations.
// ---------------------------------------------------------------------------

typedef __attribute__((ext_vector_type(2))) float v2f;
typedef __attribute__((ext_vector_type(8))) float v8f;

#define BB 16
#define NN 2048
#define DD 128
#define NSLOT 8
#define EPSV 1e-8f

// ---------------- LN row statistics (mean, rstd) ---------------------------
__global__ void ln_stats(const float* __restrict__ x, float* __restrict__ mean,
                         float* __restrict__ rstd, int C) {
  const int row = blockIdx.x;
  const int tid = threadIdx.x;                 // blockDim = 128
  const float* xr = x + (size_t)row * C;
  float s = 0.f, s2 = 0.f;
  for (int c = tid; c < C; c += 128) { float v = xr[c]; s += v; s2 += v * v; }
  __shared__ float sh[128], sh2[128];
  sh[tid] = s; sh2[tid] = s2; __syncthreads();
  for (int o = 64; o > 0; o >>= 1) {
    if (tid < o) { sh[tid] += sh[tid + o]; sh2[tid] += sh2[tid + o]; }
    __syncthreads();
  }
  if (tid == 0) {
    float m = sh[0] / C;
    float v = sh2[0] / C - m * m;
    mean[row] = m;
    rstd[row] = rsqrtf(v + 1e-5f);
  }
}

// ---------------- full LayerNorm (writes normalized rows) ------------------
__global__ void ln_rows(const float* __restrict__ x, const float* __restrict__ g,
                        const float* __restrict__ b, float* __restrict__ y, int C) {
  const int row = blockIdx.x;
  const int tid = threadIdx.x;                 // blockDim = 128
  const float* xr = x + (size_t)row * C;
  float* yr = y + (size_t)row * C;
  float s = 0.f, s2 = 0.f;
  for (int c = tid; c < C; c += 128) { float v = xr[c]; s += v; s2 += v * v; }
  __shared__ float sh[128], sh2[128];
  sh[tid] = s; sh2[tid] = s2; __syncthreads();
  for (int o = 64; o > 0; o >>= 1) {
    if (tid < o) { sh[tid] += sh[tid + o]; sh2[tid] += sh2[tid + o]; }
    __syncthreads();
  }
  const float m = sh[0] / C;
  const float var = sh2[0] / C - m * m;
  const float rs = rsqrtf(var + 1e-5f);
  for (int c = tid; c < C; c += 128)
    yr[c] = (xr[c] - m) * rs * g[c] + b[c];
}

// ---------------- generic WMMA GEMM: C = act(A @ W^T + bias) ---------------
// A: (M,K) row-major with leading dim lda.  W: (N,K) row-major.  C: ldc = N.
// One wave per 16x16 output tile. grid = (M/16, N/16), block = 32.
__global__ __launch_bounds__(32)
void gemm_tn(const float* __restrict__ A, int lda,
             const float* __restrict__ W,
             const float* __restrict__ bias,
             float* __restrict__ C, int ldc,
             int K, int relu) {
  const int m0 = blockIdx.x * 16;
  const int n0 = blockIdx.y * 16;
  const int lane = threadIdx.x;
  const int half = lane >> 4;                  // 0: K+0/1, 1: K+2/3
  const int l16  = lane & 15;
  const float* ar = A + (size_t)(m0 + l16) * lda + half * 2;
  const float* wr = W + (size_t)(n0 + l16) * K + half * 2;
  v8f acc = {};
  for (int k = 0; k < K; k += 4) {
    v2f a; a[0] = ar[k]; a[1] = ar[k + 1];
    v2f b; b[0] = wr[k]; b[1] = wr[k + 1];
    acc = __builtin_amdgcn_wmma_f32_16x16x4_f32(false, a, false, b,
                                                (short)0, acc, false, false);
  }
  const int col = n0 + l16;
  const float bv = bias ? bias[col] : 0.f;
  for (int r = 0; r < 8; ++r) {
    float val = acc[r] + bv;
    if (relu) val = fmaxf(val, 0.f);
    C[(size_t)(m0 + half * 8 + r) * ldc + col] = val;
  }
}

// ---------------- fused LN + GEMM for k = LN(x) @ Wk^T ---------------------
__global__ __launch_bounds__(32)
void gemm_ln_tn(const float* __restrict__ X, const float* __restrict__ mean,
                const float* __restrict__ rstd, const float* __restrict__ g,
                const float* __restrict__ bln,
                const float* __restrict__ W, float* __restrict__ C) {
  const int m0 = blockIdx.x * 16;
  const int n0 = blockIdx.y * 16;
  const int lane = threadIdx.x;
  const int half = lane >> 4;
  const int l16  = lane & 15;
  const int row = m0 + l16;
  const float mu = mean[row], rs = rstd[row];
  const float* xr = X + (size_t)row * DD + half * 2;
  const float* wr = W + (size_t)(n0 + l16) * DD + half * 2;
  v8f acc = {};
  for (int k = 0; k < DD; k += 4) {
    const int k0 = k + half * 2;
    v2f a;
    a[0] = (xr[k]     - mu) * rs * g[k0]     + bln[k0];
    a[1] = (xr[k + 1] - mu) * rs * g[k0 + 1] + bln[k0 + 1];
    v2f b; b[0] = wr[k]; b[1] = wr[k + 1];
    acc = __builtin_amdgcn_wmma_f32_16x16x4_f32(false, a, false, b,
                                                (short)0, acc, false, false);
  }
  for (int r = 0; r < 8; ++r)
    C[(size_t)(m0 + half * 8 + r) * DD + n0 + l16] = acc[r];
}

// ---------------- per-batch WMMA:  out[i,d] = sum_n attn[i,n] * f(v[n,d]) --
// A = gammas scaled by 1/gsum (rows 8..15 zero), B = v (or v^2).
// grid = (D/16, B, 2: {v, v^2}), block = 32.
__global__ __launch_bounds__(32)
void gemm_attn(const float* __restrict__ gammas, const float* __restrict__ gsum,
               const float* __restrict__ v, float* __restrict__ outv,
               float* __restrict__ outv2) {
  const int d0 = blockIdx.x * 16;
  const int b  = blockIdx.y;
  const int sq = blockIdx.z;
  const int lane = threadIdx.x;
  const int half = lane >> 4;
  const int l16  = lane & 15;
  const int arow  = (l16 < NSLOT) ? l16 : (NSLOT - 1);      // clamp for safety
  const float scl = (l16 < NSLOT) ? (1.f / gsum[b * NSLOT + l16]) : 0.f;
  const float* ga = gammas + ((size_t)(b * NSLOT + arow)) * NN + half * 2;
  const float* vb = v + ((size_t)b * NN + half * 2) * DD + d0 + l16;
  v8f acc = {};
  for (int n = 0; n < NN; n += 4) {
    v2f a; a[0] = ga[n] * scl; a[1] = ga[n + 1] * scl;
    float t0 = vb[(size_t)n * DD];
    float t1 = vb[(size_t)(n + 1) * DD];
    if (sq) { t0 *= t0; t1 *= t1; }
    v2f bmat; bmat[0] = t0; bmat[1] = t1;
    acc = __builtin_amdgcn_wmma_f32_16x16x4_f32(false, a, false, bmat,
                                                (short)0, acc, false, false);
  }
  if (half == 0) {                              // only rows 0..7 are real slots
    float* out = sq ? outv2 : outv;
    for (int r = 0; r < 8; ++r)
      out[((size_t)(b * NSLOT + r)) * DD + d0 + l16] = acc[r];
  }
}

// ---------------- slot init: slots = mu + exp(ls)*noise; pi = 1/NS ---------
__global__ void slots_init(const float* __restrict__ mu, const float* __restrict__ ls,
                           const float* __restrict__ noise, float* __restrict__ slots,
                           float* __restrict__ pi) {
  const int idx = blockIdx.x * 256 + threadIdx.x;   // total 16*8*256 = 32768
  const int c = idx & 255;
  slots[idx] = mu[c] + __expf(ls[c]) * noise[idx];
  if (idx < BB * NSLOT) pi[idx] = 1.0f / NSLOT;
}

// ---------------- per-slot quadratic coefficients --------------------------
// invvar = exp(-2 q_ls); beta = -2*invvar*q_mu; c0 = sum_d invvar*q_mu^2
__global__ void coeff_kernel(const float* __restrict__ qmu, const float* __restrict__ qls,
                             float* __restrict__ invvar, float* __restrict__ beta,
                             float* __restrict__ c0) {
  const int row = blockIdx.x;                  // 0..127
  const int d = threadIdx.x;                   // 0..127
  const float m = qmu[row * DD + d];
  const float iv = __expf(-2.f * qls[row * DD + d]);
  invvar[row * DD + d] = iv;
  beta[row * DD + d] = -2.f * iv * m;
  __shared__ float sh[128];
  sh[d] = iv * m * m; __syncthreads();
  for (int o = 64; o > 0; o >>= 1) {
    if (d < o) sh[d] += sh[d + o];
    __syncthreads();
  }
  if (d == 0) c0[row] = sh[0];
}

__global__ void zero128(float* __restrict__ p) { p[threadIdx.x] = 0.f; }

// ---------------- dots -> gammas (slot-softmax) + gsum reduction -----------
__global__ void dots_kernel(const float* __restrict__ k, const float* __restrict__ invvar,
                            const float* __restrict__ beta, const float* __restrict__ c0,
                            const float* __restrict__ pi, float* __restrict__ gammas,
                            float* __restrict__ gsum) {
  const int b = blockIdx.y;
  const int n = blockIdx.x * 256 + threadIdx.x;
  const float* krow = k + ((size_t)b * NN + n) * DD;
  float de[NSLOT];
  float tot = 0.f;
  for (int i = 0; i < NSLOT; ++i) {
    const float* iv = invvar + (b * NSLOT + i) * DD;
    const float* be = beta   + (b * NSLOT + i) * DD;
    float dot = c0[b * NSLOT + i];
    #pragma unroll 8
    for (int d = 0; d < DD; ++d) {
      const float kv = krow[d];
      dot = fmaf(kv * kv, iv[d], dot);
      dot = fmaf(kv, be[d], dot);
    }
    const float e = (__expf(-dot) + EPSV) * pi[b * NSLOT + i];
    de[i] = e; tot += e;
  }
  const float r = 1.f / tot;
  __shared__ float red[NSLOT * 256];
  for (int i = 0; i < NSLOT; ++i) {
    const float gm = de[i] * r;
    gammas[((size_t)(b * NSLOT + i)) * NN + n] = gm;
    red[i * 256 + threadIdx.x] = gm;
  }
  __syncthreads();
  if (threadIdx.x < NSLOT) {
    float s = 0.f;
    for (int j = 0; j < 256; ++j) s += red[threadIdx.x * 256 + j];
    atomicAdd(&gsum[b * NSLOT + threadIdx.x], s);
  }
}

// ---------------- GRU gate combine -----------------------------------------
__global__ void gru_combine(const float* __restrict__ gi, const float* __restrict__ gh,
                            const float* __restrict__ s, float* __restrict__ out) {
  const int idx = blockIdx.x * 256 + threadIdx.x;     // < 16384
  const int row = idx >> 7, d = idx & 127;
  const float ir  = gi[row * 384 + d],       hr = gh[row * 384 + d];
  const float iz  = gi[row * 384 + 128 + d], hz = gh[row * 384 + 128 + d];
  const float inn = gi[row * 384 + 256 + d], hn = gh[row * 384 + 256 + d];
  const float r = 1.f / (1.f + __expf(-(ir + hr)));
  const float z = 1.f / (1.f + __expf(-(iz + hz)));
  const float nv = tanhf(inn + r * hn);
  const float h = s[row * 256 + d];                   // mu_s lives in s[:, :128]
  out[idx] = (1.f - z) * nv + z * h;
}

// ---------------- finalize step: upd, upd_ls -> slots; pi <- gsum ----------
__global__ void finalize_step(const float* __restrict__ updgru, const float* __restrict__ t2,
                              const float* __restrict__ uv, const float* __restrict__ uv2,
                              const float* __restrict__ gsum, float* __restrict__ slots,
                              float* __restrict__ pi) {
  const int idx = blockIdx.x * 256 + threadIdx.x;     // < 16384
  const int row = idx >> 7, d = idx & 127;
  const float u = updgru[idx] + t2[idx];
  const float m1 = uv[idx];
  // sum_n attn*(v-u)^2 + EPS*sum_n attn = attn@v^2 - 2u*(attn@v) + u^2 + EPS
  const float ls = 0.5f * __logf(uv2[idx] - 2.f * u * m1 + u * u + EPSV);
  slots[row * 256 + d] = u;
  slots[row * 256 + 128 + d] = ls;
  if (d == 0) pi[row] = gsum[row];
}

// ===========================================================================
extern "C" void kernel_launch(void* const* d_in, const int* in_sizes, int n_in,
                              void* d_out, int out_size, void* d_ws, size_t ws_size,
                              hipStream_t stream) {
  const float* inputs = (const float*)d_in[0];
  const float* init_noise = (const float*)d_in[1];
  const float* slots_mu = (const float*)d_in[2];
  const float* slots_logsigma = (const float*)d_in[3];
  const float* Wq  = (const float*)d_in[4];
  const float* Wk  = (const float*)d_in[5];
  const float* gWih = (const float*)d_in[6];
  const float* gWhh = (const float*)d_in[7];
  const float* gbih = (const float*)d_in[8];
  const float* gbhh = (const float*)d_in[9];
  const float* mW1 = (const float*)d_in[10];
  const float* mb1 = (const float*)d_in[11];
  const float* mW2 = (const float*)d_in[12];
  const float* mb2 = (const float*)d_in[13];
  const float* ling = (const float*)d_in[14];
  const float* linb = (const float*)d_in[15];
  const float* lsg = (const float*)d_in[16];
  const float* lsb = (const float*)d_in[17];
  const float* lmg = (const float*)d_in[18];
  const float* lmb = (const float*)d_in[19];
  const float* oW1 = (const float*)d_in[20];
  const float* ob1 = (const float*)d_in[21];
  const float* oW2 = (const float*)d_in[22];
  const float* ob2 = (const float*)d_in[23];

  float* w = (float*)d_ws;
  float* kbuf  = w; w += (size_t)BB * NN * DD;   // 4.19M
  float* rmean = w; w += BB * NN;
  float* rrstd = w; w += BB * NN;
  float* slots = w; w += BB * NSLOT * 256;
  float* pi    = w; w += BB * NSLOT;
  float* s     = w; w += BB * NSLOT * 256;
  float* q_mu  = w; w += BB * NSLOT * DD;
  float* q_ls  = w; w += BB * NSLOT * DD;
  float* invvar= w; w += BB * NSLOT * DD;
  float* beta  = w; w += BB * NSLOT * DD;
  float* c0    = w; w += BB * NSLOT;
  float* gammas= w; w += (size_t)BB * NSLOT * NN;
  float* gsum  = w; w += BB * NSLOT;
  float* uv    = w; w += BB * NSLOT * DD;
  float* uv2   = w; w += BB * NSLOT * DD;
  float* gi    = w; w += 128 * 384;
  float* gh    = w; w += 128 * 384;
  float* updg  = w; w += 128 * 128;
  float* hln   = w; w += 128 * 128;
  float* t1b   = w; w += 128 * 128;
  float* t2b   = w; w += 128 * 128;
  float* hcat  = w; w += 128 * 256;

  // --- k = LN(inputs) @ Wk^T  (fused LN, WMMA) ---
  ln_stats<<<BB * NN, 128, 0, stream>>>(inputs, rmean, rrstd, DD);
  gemm_ln_tn<<<dim3(BB * NN / 16, DD / 16), 32, 0, stream>>>(
      inputs, rmean, rrstd, ling, linb, Wk, kbuf);

  // --- slots = mu + exp(logsigma)*noise; pi = 1/NS ---
  slots_init<<<128, 256, 0, stream>>>(slots_mu, slots_logsigma, init_noise, slots, pi);

  for (int it = 0; it < 4; ++it) {             // ITERS=3 + stop_gradient step
    ln_rows<<<128, 128, 0, stream>>>(slots, lsg, lsb, s, 256);
    // q_mu / q_ls = (mu_s | ls_s) @ Wq^T
    gemm_tn<<<dim3(8, 8), 32, 0, stream>>>(s,       256, Wq, nullptr, q_mu, DD, DD, 0);
    gemm_tn<<<dim3(8, 8), 32, 0, stream>>>(s + 128, 256, Wq, nullptr, q_ls, DD, DD, 0);
    coeff_kernel<<<128, 128, 0, stream>>>(q_mu, q_ls, invvar, beta, c0);
    zero128<<<1, 128, 0, stream>>>(gsum);
    dots_kernel<<<dim3(NN / 256, BB), 256, 0, stream>>>(kbuf, invvar, beta, c0, pi,
                                                        gammas, gsum);
    // upd_v = attn @ v,  upd_v2 = attn @ v^2   (WMMA, M padded 8->16)
    gemm_attn<<<dim3(DD / 16, BB, 2), 32, 0, stream>>>(gammas, gsum, kbuf, uv, uv2);
    // GRU gates
    gemm_tn<<<dim3(8, 24), 32, 0, stream>>>(uv, 128, gWih, gbih, gi, 384, DD, 0);
    gemm_tn<<<dim3(8, 24), 32, 0, stream>>>(s,  256, gWhh, gbhh, gh, 384, DD, 0);
    gru_combine<<<64, 256, 0, stream>>>(gi, gh, s, updg);
    // residual MLP on mu
    ln_rows<<<128, 128, 0, stream>>>(updg, lmg, lmb, hln, 128);
    gemm_tn<<<dim3(8, 8), 32, 0, stream>>>(hln, 128, mW1, mb1, t1b, 128, DD, 1);
    gemm_tn<<<dim3(8, 8), 32, 0, stream>>>(t1b, 128, mW2, mb2, t2b, 128, 128, 0);
    finalize_step<<<64, 256, 0, stream>>>(updg, t2b, uv, uv2, gsum, slots, pi);
  }

  // --- output head: relu(slots @ oW1^T + b1) @ oW2^T + b2 ---
  gemm_tn<<<dim3(8, 16), 32, 0, stream>>>(slots, 256, oW1, ob1, hcat, 256, 256, 1);
  gemm_tn<<<dim3(8, 8), 32, 0, stream>>>(hcat, 256, oW2, ob2, (float*)d_out, 128, 256, 0);
}